// ContourLoss_37726992728108
// MI455X (gfx1250) — compile-verified
//
#include <hip/hip_runtime.h>
#include <hip/hip_bf16.h>

// ---------------------------------------------------------------------------
// ContourLoss for MI455X (gfx1250, wave32).
// Core trick: cross(seg_j, p) = [p.x, p.y, 1, 0] . [-dy_j, dx_j, c_j, 0]
// => all four d1..d4 matrices of the intersection term are 16x16 tiles of
//    M x 4 by 4 x M f32 matmuls -> V_WMMA_F32_16X16X4_F32 (exact f32 math).
// The transposed accesses (d3, d4) are produced directly by swapping the
// A/B roles, so no cross-lane transpose is needed.
// ---------------------------------------------------------------------------

#define BATCH   8
#define SEQ     2048
#define PSTR    2064                  // padded per-sample point stride (>= 2049)
#define TILES   128                   // ceil((SEQ-1)/16)
#define NPAIRS  (TILES * (TILES + 1) / 2)   // 8256 tile-pairs (J >= I)
#define TOTTASK (BATCH * NPAIRS)

typedef __attribute__((ext_vector_type(2))) float v2f;
typedef __attribute__((ext_vector_type(8))) float v8f;

static __device__ __forceinline__ v8f wmma4(v2f a, v2f b) {
  v8f c = {};
  // D(16x16,f32) = A(16x4,f32) x B(4x16,f32) + 0
  return __builtin_amdgcn_wmma_f32_16x16x4_f32(
      /*neg_a=*/false, a, /*neg_b=*/false, b,
      /*c_mod=*/(short)0, c, /*reuse_a=*/false, /*reuse_b=*/false);
}

// ---------------------------------------------------------------------------
// Kernel 1: per-sample (8 blocks x 256 threads).
//  - cls / point / orient partial reductions -> blockPartials[b*5 + {0..4}]
//  - stable compaction of non-none points -> ptsX/ptsY (zero padded), nArr[b]
// ---------------------------------------------------------------------------
__global__ __launch_bounds__(256) void contour_prep(
    const float* __restrict__ point_pred, const float* __restrict__ orient_pred,
    const float* __restrict__ class_pred, const float* __restrict__ target_seq,
    const int* __restrict__ padding_mask, float* __restrict__ ptsX,
    float* __restrict__ ptsY, int* __restrict__ nArr,
    float* __restrict__ blockPartials) {
  const int b = blockIdx.x;
  const int t = threadIdx.x;
  __shared__ int sCnt[256];
  __shared__ float sRed[256];

  // zero-fill padded point arrays (overwritten by compaction below)
  for (int idx = t; idx < PSTR; idx += 256) {
    ptsX[b * PSTR + idx] = 0.0f;
    ptsY[b * PSTR + idx] = 0.0f;
  }

  float clsSum = 0.f, validCnt = 0.f, ptSum = 0.f, orSum = 0.f, nnCnt = 0.f;
  float pxr[8], pyr[8];
  int flg = 0;

#pragma unroll
  for (int k = 0; k < 8; ++k) {
    const int s = t * 8 + k;
    const int base = b * SEQ + s;
    const bool valid = (padding_mask[base] == 0);
    const float* ts = target_seq + (size_t)base * 5;
    const int tc = (int)ts[4];
    const float* cp = class_pred + (size_t)base * 4;
    const float x0 = cp[0], x1 = cp[1], x2 = cp[2], x3 = cp[3];
    const float mx = fmaxf(fmaxf(x0, x1), fmaxf(x2, x3));
    const float lse = mx + logf(expf(x0 - mx) + expf(x1 - mx) +
                                expf(x2 - mx) + expf(x3 - mx));
    const float xt = (tc == 0) ? x0 : ((tc == 1) ? x1 : ((tc == 2) ? x2 : x3));
    if (valid) { clsSum += lse - xt; validCnt += 1.0f; }

    const bool nn = valid && (tc != 0);
    const float px = point_pred[(size_t)base * 2 + 0];
    const float py = point_pred[(size_t)base * 2 + 1];
    pxr[k] = px; pyr[k] = py;
    flg |= (nn ? 1 : 0) << k;
    if (nn) {
      const float ddx = (px - ts[0]) * (1.0f / 224.0f);
      const float ddy = (py - ts[1]) * (1.0f / 224.0f);
      const float ax = fabsf(ddx), ay = fabsf(ddy);
      const float fx = (ax < 1.0f) ? 0.5f * ddx * ddx : ax - 0.5f;
      const float fy = (ay < 1.0f) ? 0.5f * ddy * ddy : ay - 0.5f;
      ptSum += 0.5f * (fx + fy);
      const float ox = orient_pred[(size_t)base * 2 + 0];
      const float oy = orient_pred[(size_t)base * 2 + 1];
      orSum += 1.0f - (ox * ts[2] + oy * ts[3]);
      nnCnt += 1.0f;
    }
  }

  // stable compaction: inclusive Hillis-Steele scan over per-thread counts
  sCnt[t] = __popc(flg);
  __syncthreads();
  for (int off = 1; off < 256; off <<= 1) {
    const int v = sCnt[t];
    const int add = (t >= off) ? sCnt[t - off] : 0;
    __syncthreads();
    sCnt[t] = v + add;
    __syncthreads();
  }
  const int excl = sCnt[t] - __popc(flg);
  const int total = sCnt[255];
  if (t == 0) nArr[b] = total;

  int wr = excl;
#pragma unroll
  for (int k = 0; k < 8; ++k) {
    if ((flg >> k) & 1) {
      ptsX[b * PSTR + wr] = pxr[k];
      ptsY[b * PSTR + wr] = pyr[k];
      ++wr;
    }
  }

  // deterministic block reductions of the 5 partials
  float vals[5] = {clsSum, validCnt, ptSum, orSum, nnCnt};
#pragma unroll
  for (int i = 0; i < 5; ++i) {
    __syncthreads();
    sRed[t] = vals[i];
    __syncthreads();
    for (int ofs = 128; ofs > 0; ofs >>= 1) {
      if (t < ofs) sRed[t] += sRed[t + ofs];
      __syncthreads();
    }
    if (t == 0) blockPartials[b * 5 + i] = sRed[0];
  }
}

// ---------------------------------------------------------------------------
// Kernel 2: one wave per (sample, tile-pair) task, grid-stride. 4 WMMAs/task.
// ---------------------------------------------------------------------------
__global__ __launch_bounds__(256) void contour_pairs(
    const float* __restrict__ ptsX, const float* __restrict__ ptsY,
    const int* __restrict__ nArr, float* __restrict__ pairSum,
    float* __restrict__ pairCnt, int totalWaves) {
  const int lane = threadIdx.x & 31;
  const int waveId = (int)((blockIdx.x * blockDim.x + threadIdx.x) >> 5);
  const bool lo = lane < 16;
  const int r16 = lane & 15;

  float accS = 0.0f;
  float accC = 0.0f;

  for (int task = waveId; task < TOTTASK; task += totalWaves) {
    const int b = task / NPAIRS;
    const int p = task - b * NPAIRS;
    // decode p = J*(J+1)/2 + I  (I <= J), sqrt + integer fixup
    int J = (int)((sqrtf(8.0f * (float)p + 1.0f) - 1.0f) * 0.5f);
    while ((J + 1) * (J + 2) / 2 <= p) ++J;
    while (J * (J + 1) / 2 > p) --J;
    const int I = p - J * (J + 1) / 2;

    const int n = nArr[b];            // wave-uniform
    const int nseg = n - 1;
    if (n < 4) continue;              // mask includes (n >= 4)
    if (J * 16 >= nseg) continue;     // whole tile-pair masked (J >= I)

    const float* bx = ptsX + b * PSTR;
    const float* by = ptsY + b * PSTR;

    // per-lane row/col values for tiles I and J (lanes 0-15 / 16-31 mirror)
    const int iI = I * 16 + r16;
    const float sxI = bx[iI], syI = by[iI];
    const float exI = bx[iI + 1], eyI = by[iI + 1];
    const float dxI = exI - sxI, dyI = eyI - syI;
    const float cI = dyI * sxI - dxI * syI;

    const int iJ = J * 16 + r16;
    const float sxJ = bx[iJ], syJ = by[iJ];
    const float exJ = bx[iJ + 1], eyJ = by[iJ + 1];
    const float dxJ = exJ - sxJ, dyJ = eyJ - syJ;
    const float cJ = dyJ * sxJ - dxJ * syJ;

    // Fragments per ISA layout: VGPR0 = K0 (lanes 0-15) / K2 (lanes 16-31),
    //                           VGPR1 = K1 / K3. Same layout for A and B here.
    v2f stI; stI.x = lo ? sxI : 1.0f;  stI.y = lo ? syI : 0.0f;
    v2f enI; enI.x = lo ? exI : 1.0f;  enI.y = lo ? eyI : 0.0f;
    v2f sgI; sgI.x = lo ? -dyI : cI;   sgI.y = lo ? dxI : 0.0f;
    v2f stJ; stJ.x = lo ? sxJ : 1.0f;  stJ.y = lo ? syJ : 0.0f;
    v2f enJ; enJ.x = lo ? exJ : 1.0f;  enJ.y = lo ? eyJ : 0.0f;
    v2f sgJ; sgJ.x = lo ? -dyJ : cJ;   sgJ.y = lo ? dxJ : 0.0f;

    // d1[i,j] = start_i . seg_j    d2[i,j] = end_i . seg_j
    // d3[i,j] = seg_i . start_j    d4[i,j] = seg_i . end_j   (pre-transposed)
    const v8f D1  = wmma4(stI, sgJ);
    const v8f D2  = wmma4(enI, sgJ);
    const v8f D1t = wmma4(sgI, stJ);
    const v8f D2t = wmma4(sgI, enJ);

    const int jcol = J * 16 + r16;    // C layout: N = lane%16
#pragma unroll
    for (int r = 0; r < 8; ++r) {
      const int irow = I * 16 + r + (lo ? 0 : 8);  // C layout: M = r (+8 hi)
      const bool msk = (jcol >= irow + 2) && (irow < nseg) && (jcol < nseg) &&
                       !((irow == 0) && (jcol == nseg - 1));
      const float q1 = 0.01f * D1[r] * D2[r];
      const float q2 = 0.01f * D1t[r] * D2t[r];
      // sigmoid(-q) = 1 / (1 + e^q)
      const float sv = (1.0f / (1.0f + __expf(q1))) *
                       (1.0f / (1.0f + __expf(q2)));
      accS += msk ? sv : 0.0f;
      accC += msk ? 1.0f : 0.0f;
    }
  }

  // wave32 reduction, lane 0 writes this wave's deterministic slot
#pragma unroll
  for (int off = 16; off > 0; off >>= 1) {
    accS += __shfl_down(accS, off, 32);
    accC += __shfl_down(accC, off, 32);
  }
  if (lane == 0) {
    pairSum[waveId] = accS;
    pairCnt[waveId] = accC;
  }
}

// ---------------------------------------------------------------------------
// Kernel 3: single-block deterministic finalize -> 5 outputs
// ---------------------------------------------------------------------------
__global__ __launch_bounds__(256) void contour_finalize(
    const float* __restrict__ blockPartials, const float* __restrict__ pairSum,
    const float* __restrict__ pairCnt, int nWaves, float* __restrict__ out) {
  __shared__ float sA[256];
  __shared__ float sB[256];
  float a = 0.f, c = 0.f;
  for (int i = threadIdx.x; i < nWaves; i += 256) {
    a += pairSum[i];
    c += pairCnt[i];
  }
  sA[threadIdx.x] = a;
  sB[threadIdx.x] = c;
  __syncthreads();
  for (int ofs = 128; ofs > 0; ofs >>= 1) {
    if (threadIdx.x < ofs) {
      sA[threadIdx.x] += sA[threadIdx.x + ofs];
      sB[threadIdx.x] += sB[threadIdx.x + ofs];
    }
    __syncthreads();
  }
  if (threadIdx.x == 0) {
    float clsSum = 0.f, validCnt = 0.f, ptSum = 0.f, orSum = 0.f, nnCnt = 0.f;
    for (int b = 0; b < BATCH; ++b) {
      clsSum   += blockPartials[b * 5 + 0];
      validCnt += blockPartials[b * 5 + 1];
      ptSum    += blockPartials[b * 5 + 2];
      orSum    += blockPartials[b * 5 + 3];
      nnCnt    += blockPartials[b * 5 + 4];
    }
    const float cls = clsSum / fmaxf(validCnt, 1.0f);
    const float den = fmaxf(nnCnt, 1.0f);
    const float pt = ptSum / den;
    const float orn = orSum / den;
    const float isum = sA[0], icnt = sB[0];
    const float inter = (icnt > 0.0f) ? isum / fmaxf(icnt, 1.0f) : 0.0f;
    out[0] = 1.0f * pt + 0.5f * orn + 1.0f * cls + 0.1f * inter;
    out[1] = pt;
    out[2] = orn;
    out[3] = cls;
    out[4] = inter;
  }
}

extern "C" void kernel_launch(void* const* d_in, const int* in_sizes, int n_in,
                              void* d_out, int out_size, void* d_ws,
                              size_t ws_size, hipStream_t stream) {
  const float* point_pred   = (const float*)d_in[0];
  const float* orient_pred  = (const float*)d_in[1];
  const float* class_pred   = (const float*)d_in[2];
  const float* target_seq   = (const float*)d_in[3];
  const int*   padding_mask = (const int*)d_in[4];

  char* ws = (char*)d_ws;
  // layout (bytes): ptsX[8*2064 f32] | ptsY | nArr[8 i32] | blockPartials[40 f32]
  //                 | pairSum[4096 f32] | pairCnt[4096 f32]   (~165 KB total)
  float* ptsX          = (float*)(ws + 0);
  float* ptsY          = (float*)(ws + 66048);
  int*   nArr          = (int*)  (ws + 132096);
  float* blockPartials = (float*)(ws + 132352);
  float* pairSum       = (float*)(ws + 132608);
  float* pairCnt       = (float*)(ws + 148992);
  float* out           = (float*)d_out;

  contour_prep<<<BATCH, 256, 0, stream>>>(point_pred, orient_pred, class_pred,
                                          target_seq, padding_mask, ptsX, ptsY,
                                          nArr, blockPartials);

  const int nBlocks = 512;
  const int nWaves = nBlocks * 256 / 32;  // 4096 waves, ~16 tile-pair tasks each
  contour_pairs<<<nBlocks, 256, 0, stream>>>(ptsX, ptsY, nArr, pairSum,
                                             pairCnt, nWaves);

  contour_finalize<<<1, 256, 0, stream>>>(blockPartials, pairSum, pairCnt,
                                          nWaves, out);
}